// Mamba_5480378270276
// MI455X (gfx1250) — compile-verified
//
#include <hip/hip_runtime.h>
#include <hip/hip_bf16.h>
#include <math.h>

// Problem constants (match reference)
#define B_  2
#define L_  1024
#define DM  768
#define DI  1536
#define NS  16
#define RK  48
#define KC  4
#define BL  (B_ * L_)          // 2048 rows
#define XDB (RK + 2 * NS)      // 80

typedef __attribute__((ext_vector_type(2))) float v2f;
typedef __attribute__((ext_vector_type(8))) float v8f;

// ---------------------------------------------------------------------------
// Async global -> LDS copy of 16 bytes per lane (GVS mode: 64-bit SGPR base +
// 32-bit per-lane byte offset).  Tracked by ASYNCcnt.
// ---------------------------------------------------------------------------
__device__ __forceinline__ void async_copy_b128(uint32_t lds_byte_off,
                                                uint32_t glb_byte_off,
                                                uint64_t base)
{
    asm volatile("global_load_async_to_lds_b128 %0, %1, %2"
                 :
                 : "v"(lds_byte_off), "v"(glb_byte_off), "s"(base)
                 : "memory");
}

__device__ __forceinline__ void wait_async0()
{
    asm volatile("s_wait_asynccnt 0x0" ::: "memory");
}

__device__ __forceinline__ uint32_t lds_off_of(const void* p)
{
    // Flat shared-aperture pointer: low 32 bits == LDS byte offset.
    return (uint32_t)(uintptr_t)p;
}

// ---------------------------------------------------------------------------
// fp32 WMMA GEMM:  C[M x N] = A[M x K] (row stride lda) * W[N x K]^T
// (row stride ldw), C row stride ldc.  V_WMMA_F32_16X16X4_F32.
//
// Block = 256 threads = 8 waves, block tile 64x64, K chunk = 16.
// A/W chunk tiles are staged in LDS with double-buffered async-to-LDS b128
// copies; each wave owns two 16x16 C tiles (one A fragment feeds 2 WMMAs).
// Fragment layout per ISA 7.12.2:
//   A 16x4 : lanes 0-15 -> M=lane, vgpr{0,1}=K{0,1}; lanes 16-31 -> K{2,3}
//   B 4x16 : lanes 0-15 -> N=lane, vgpr{0,1}=K{0,1}; lanes 16-31 -> K{2,3}
//   C      : vgpr j -> row (j + 8*half), col = lane&15
// Requirements: M % 64 == 0, K % 16 == 0, N % 16 == 0 (N tiles guarded).
// ---------------------------------------------------------------------------
__global__ __launch_bounds__(256) void gemm_f32_wmma(
    const float* __restrict__ A, int lda,
    const float* __restrict__ W, int ldw,
    float* __restrict__ C, int ldc,
    int M, int N, int K)
{
    __shared__ float As[2][64][16];
    __shared__ float Ws[2][64][16];

    const int m0 = blockIdx.y * 64;
    const int n0 = blockIdx.x * 64;

    // Cooperative staging role: 256 threads cover 64 rows x 16 floats,
    // 4 threads per row, 4 floats (16 B) per thread.
    const int tr = threadIdx.x >> 2;        // 0..63 row within tile
    const int tc = (threadIdx.x & 3) * 4;   // 0,4,8,12 float col within chunk
    const int ga_row = m0 + tr;                              // M % 64 == 0
    const int gw_row = (n0 + tr < N) ? (n0 + tr) : (N - 1);  // clamp (no OOB)

    const uint64_t baseA = (uint64_t)(uintptr_t)A;
    const uint64_t baseW = (uint64_t)(uintptr_t)W;

    // Per-wave tile assignment
    const int lane = threadIdx.x & 31;
    const int wave = threadIdx.x >> 5;      // 0..7
    const int half = lane >> 4;
    const int lid  = lane & 15;
    const int koff = half * 2;
    const int mt   = wave >> 1;             // 0..3  -> m tile
    const int ntl  = (wave & 1) * 32;       // local n of first tile (0 or 32)
    const int mrow = mt * 16 + lid;         // A fragment row in LDS tile

    v8f c0 = {};
    v8f c1 = {};

    const int nchunks = K >> 4;

    // Prologue: stage chunk 0 into buffer 0
    {
        const uint32_t ga = (uint32_t)((size_t)ga_row * lda + tc) * 4u;
        const uint32_t gw = (uint32_t)((size_t)gw_row * ldw + tc) * 4u;
        async_copy_b128(lds_off_of(&As[0][tr][tc]), ga, baseA);
        async_copy_b128(lds_off_of(&Ws[0][tr][tc]), gw, baseW);
    }

    for (int cidx = 0; cidx < nchunks; ++cidx) {
        const int cur = cidx & 1;
        wait_async0();
        __syncthreads();                    // buffer 'cur' fully staged

        if (cidx + 1 < nchunks) {           // stage next chunk into other buf
            const int k0 = (cidx + 1) << 4;
            const uint32_t ga = (uint32_t)((size_t)ga_row * lda + k0 + tc) * 4u;
            const uint32_t gw = (uint32_t)((size_t)gw_row * ldw + k0 + tc) * 4u;
            async_copy_b128(lds_off_of(&As[1 - cur][tr][tc]), ga, baseA);
            async_copy_b128(lds_off_of(&Ws[1 - cur][tr][tc]), gw, baseW);
        }

#pragma unroll
        for (int kk = 0; kk < 16; kk += 4) {
            v2f a  = *(const v2f*)&As[cur][mrow][kk + koff];
            v2f b0 = *(const v2f*)&Ws[cur][ntl + lid][kk + koff];
            v2f b1 = *(const v2f*)&Ws[cur][ntl + 16 + lid][kk + koff];
            c0 = __builtin_amdgcn_wmma_f32_16x16x4_f32(
                     false, a, false, b0, (short)0, c0, false, false);
            c1 = __builtin_amdgcn_wmma_f32_16x16x4_f32(
                     false, a, false, b1, (short)0, c1, false, false);
        }

        __syncthreads();                    // all reads of 'cur' done
    }

    // Store: tile rows m0 + mt*16 + half*8 + j, cols n_tile + lid
    const int crow = m0 + mt * 16 + half * 8;
    const int nA = n0 + ntl;
    const int nB = nA + 16;
    if (nA < N) {
        float* Cp = C + (size_t)crow * ldc + nA + lid;
#pragma unroll
        for (int j = 0; j < 8; ++j) Cp[(size_t)j * ldc] = c0[j];
    }
    if (nB < N) {
        float* Cp = C + (size_t)crow * ldc + nB + lid;
#pragma unroll
        for (int j = 0; j < 8; ++j) Cp[(size_t)j * ldc] = c1[j];
    }
}

// ---------------------------------------------------------------------------
// Depthwise causal conv1d (K=4) + SiLU over the x half of xz.
// dir=1 reads the sequence flipped (backward branch works in scan order s,
// which maps to original position l = L-1-s).
// ---------------------------------------------------------------------------
__global__ __launch_bounds__(256) void conv_silu_kernel(
    const float* __restrict__ xz,        // (B,L,2*DI)
    const float* __restrict__ cw,        // (DI,KC)
    const float* __restrict__ cb,        // (DI)
    float* __restrict__ xc,              // (B,L,DI) in scan order
    int dir)
{
    int t = blockIdx.x * blockDim.x + threadIdx.x;
    if (t >= BL * DI) return;
    int d = t % DI;
    int s = (t / DI) % L_;
    int b = t / (DI * L_);

    float acc = cb[d];
#pragma unroll
    for (int k = 0; k < KC; ++k) {
        int p = s - (KC - 1) + k;          // scan-order position
        if (p >= 0) {
            int l = dir ? (L_ - 1 - p) : p; // original position
            acc += cw[d * KC + k] * xz[((size_t)(b * L_ + l)) * (2 * DI) + d];
        }
    }
    xc[t] = acc / (1.0f + __expf(-acc));   // SiLU
}

// ---------------------------------------------------------------------------
// delta = softplus(dt_proj_out + bias);  du = delta * x_conv  (elementwise)
// delta overwrites dtp in place (same-index read/write, race-free).
// ---------------------------------------------------------------------------
__global__ __launch_bounds__(256) void delta_du_kernel(
    float* __restrict__ dtp,             // (B,L,DI) in, delta out
    const float* __restrict__ bias,      // (DI)
    const float* __restrict__ xc,        // (B,L,DI)
    float* __restrict__ du)              // (B,L,DI)
{
    int t = blockIdx.x * blockDim.x + threadIdx.x;
    if (t >= BL * DI) return;
    int d = t % DI;
    float v = dtp[t] + bias[d];
    float dl = (v > 20.0f) ? v : log1pf(__expf(v));
    dtp[t] = dl;
    du[t]  = dl * xc[t];
}

// ---------------------------------------------------------------------------
// Selective scan.  One thread per (b, d, n): 16 lanes (one per state n) form a
// group, reduce y = sum_n h*C with 4 shfl_xor steps.  Serial chain per step is
// a single exp + fma per lane.  dir=1: inputs are already in scan order; the
// gated output is written back to the flipped original position (+= onto the
// forward branch's result — kernels are stream-ordered, one writer per elem).
// ---------------------------------------------------------------------------
__global__ __launch_bounds__(256) void scan_kernel(
    const float* __restrict__ delta,     // (B,L,DI)      scan order
    const float* __restrict__ du,        // (B,L,DI)      scan order
    const float* __restrict__ xdbl,      // (B,L,80): [dt | Bc | Cc] scan order
    const float* __restrict__ xc,        // (B,L,DI)      scan order
    const float* __restrict__ xz,        // (B,L,2*DI)    original order (z half)
    const float* __restrict__ Dp,        // (DI)
    const float* __restrict__ A_log,     // (DI,NS)
    float* __restrict__ acc,             // (B,L,DI)      original order
    int dir)
{
    const int n  = threadIdx.x & 15;
    const int cl = threadIdx.x >> 4;     // 16 channels per 256-thread block
    const int d  = blockIdx.x * 16 + cl;
    const int b  = blockIdx.y;

    const float A  = -__expf(A_log[d * NS + n]);
    const float Dv = Dp[d];

    float h = 0.0f;
    const size_t rb = (size_t)b * L_;
    for (int s = 0; s < L_; ++s) {
        const size_t row = rb + s;
        const float dlt = delta[row * DI + d];
        const float duv = du[row * DI + d];
        const float Bv  = xdbl[row * XDB + RK + n];
        const float Cv  = xdbl[row * XDB + RK + NS + n];

        const float dA = __expf(dlt * A);
        h = fmaf(dA, h, duv * Bv);

        float p = h * Cv;
        p += __shfl_xor(p, 1, 32);
        p += __shfl_xor(p, 2, 32);
        p += __shfl_xor(p, 4, 32);
        p += __shfl_xor(p, 8, 32);

        if (n == 0) {
            const int l = dir ? (L_ - 1 - s) : s;
            const float y = p + xc[row * DI + d] * Dv;
            const float z = xz[((size_t)(rb + l)) * (2 * DI) + DI + d];
            const float g = y * (z / (1.0f + __expf(-z)));
            const size_t o = ((size_t)(rb + l)) * DI + d;
            if (dir) acc[o] += g;
            else     acc[o]  = g;
        }
    }
}

// ---------------------------------------------------------------------------
extern "C" void kernel_launch(void* const* d_in, const int* in_sizes, int n_in,
                              void* d_out, int out_size, void* d_ws, size_t ws_size,
                              hipStream_t stream)
{
    const float* hidden     = (const float*)d_in[0];   // (B,L,DM)
    const float* in_proj_w  = (const float*)d_in[1];   // (2*DI, DM)
    const float* conv_w_f   = (const float*)d_in[2];   // (DI,KC)
    const float* conv_b_f   = (const float*)d_in[3];   // (DI)
    const float* x_proj_f   = (const float*)d_in[4];   // (80, DI)
    const float* dt_proj_f  = (const float*)d_in[5];   // (DI, RK)
    const float* dt_bias_f  = (const float*)d_in[6];   // (DI)
    const float* A_log_f    = (const float*)d_in[7];   // (DI,NS)
    const float* D_f        = (const float*)d_in[8];   // (DI)
    const float* conv_w_b   = (const float*)d_in[9];
    const float* conv_b_b   = (const float*)d_in[10];
    const float* x_proj_b   = (const float*)d_in[11];
    const float* dt_proj_b  = (const float*)d_in[12];
    const float* dt_bias_b  = (const float*)d_in[13];
    const float* A_log_b    = (const float*)d_in[14];
    const float* D_b        = (const float*)d_in[15];
    const float* out_proj_w = (const float*)d_in[16];  // (DM, DI)
    float* out = (float*)d_out;                        // (B,L,DM)

    // Workspace partition (floats)
    float* ws   = (float*)d_ws;
    float* xz   = ws;                         // BL * 2*DI
    float* xc   = xz   + (size_t)BL * 2 * DI; // BL * DI
    float* xdbl = xc   + (size_t)BL * DI;     // BL * 80
    float* dtp  = xdbl + (size_t)BL * XDB;    // BL * DI  (dt_proj out -> delta)
    float* du   = dtp  + (size_t)BL * DI;     // BL * DI
    float* acc  = du   + (size_t)BL * DI;     // BL * DI

    const int EW_GRID = (BL * DI + 255) / 256;

    // 1) xz = hidden @ in_proj_w^T   (2048 x 3072, K=768)
    {
        dim3 g((2 * DI + 63) / 64, BL / 64);
        gemm_f32_wmma<<<g, 256, 0, stream>>>(hidden, DM, in_proj_w, DM,
                                             xz, 2 * DI, BL, 2 * DI, DM);
    }

    const float* cw[2]  = {conv_w_f,  conv_w_b};
    const float* cb[2]  = {conv_b_f,  conv_b_b};
    const float* xpw[2] = {x_proj_f,  x_proj_b};
    const float* dpw[2] = {dt_proj_f, dt_proj_b};
    const float* dpb[2] = {dt_bias_f, dt_bias_b};
    const float* Al[2]  = {A_log_f,   A_log_b};
    const float* Dd[2]  = {D_f,       D_b};

    for (int dir = 0; dir < 2; ++dir) {
        // 2) depthwise causal conv + SiLU (scan order)
        conv_silu_kernel<<<EW_GRID, 256, 0, stream>>>(xz, cw[dir], cb[dir], xc, dir);

        // 3) x_dbl = xc @ x_proj_w^T   (2048 x 80, K=1536)
        {
            dim3 g((XDB + 63) / 64, BL / 64);
            gemm_f32_wmma<<<g, 256, 0, stream>>>(xc, DI, xpw[dir], DI,
                                                 xdbl, XDB, BL, XDB, DI);
        }

        // 4) dtp = x_dbl[:, :48] @ dt_proj_w^T   (2048 x 1536, K=48, lda=80)
        {
            dim3 g((DI + 63) / 64, BL / 64);
            gemm_f32_wmma<<<g, 256, 0, stream>>>(xdbl, XDB, dpw[dir], RK,
                                                 dtp, DI, BL, DI, RK);
        }

        // 5) delta = softplus(dtp + bias); du = delta * xc
        delta_du_kernel<<<EW_GRID, 256, 0, stream>>>(dtp, dpb[dir], xc, du);

        // 6) selective scan + SiLU(z) gating; fwd stores, bwd accumulates
        {
            dim3 g(DI / 16, B_);
            scan_kernel<<<g, 256, 0, stream>>>(dtp, du, xdbl, xc, xz,
                                               Dd[dir], Al[dir], acc, dir);
        }
    }

    // 7) out = acc @ out_proj_w^T    (2048 x 768, K=1536)
    {
        dim3 g((DM + 63) / 64, BL / 64);
        gemm_f32_wmma<<<g, 256, 0, stream>>>(acc, DI, out_proj_w, DI,
                                             out, DM, BL, DM, DI);
    }
}